// OneLayerGCN_9929964389285
// MI455X (gfx1250) — compile-verified
//
#include <hip/hip_runtime.h>

typedef __attribute__((ext_vector_type(2))) float v2f;
typedef __attribute__((ext_vector_type(8))) float v8f;

#define N_NODES  131072     // B*S
#define F_INK    300
#define F_OUTK   64
#define N_EDGE   2097152
#define S_SUB    128
#define B_SUB    1024

// ---------------------------------------------------------------------------
// Kernel 1: in/out degree via float atomics (counters are L2-resident)
// ---------------------------------------------------------------------------
__global__ void gcn_degree_kernel(const int* __restrict__ src,
                                  const int* __restrict__ dst,
                                  float* __restrict__ deg_out,
                                  float* __restrict__ deg_in) {
    int i = blockIdx.x * blockDim.x + threadIdx.x;
    if (i < N_EDGE) {
        atomicAdd(&deg_out[src[i]], 1.0f);
        atomicAdd(&deg_in[dst[i]], 1.0f);
    }
}

// ---------------------------------------------------------------------------
// Kernel 2: H = (in_feat * rsqrt(deg_out)) @ W  via V_WMMA_F32_16X16X4_F32.
// 8 waves / block, one 16x16 tile per wave: 32 rows x 64 cols per block.
// Grid covers N exactly -> no divergence -> EXEC all-1s for WMMA.
// ---------------------------------------------------------------------------
__global__ void gcn_gemm_wmma_kernel(const float* __restrict__ X,
                                     const float* __restrict__ W,
                                     const float* __restrict__ deg_out,
                                     float* __restrict__ H) {
    const int lane    = threadIdx.x & 31;
    const int wave    = threadIdx.x >> 5;   // 0..7
    const int rowTile = wave & 1;           // 2 row tiles of 16
    const int colTile = wave >> 1;          // 4 col tiles of 16
    const int m       = lane & 15;
    const int half    = lane >> 4;          // K-half select (ISA A/B layout)

    const int rowBase = blockIdx.x * 32 + rowTile * 16;
    const int row     = rowBase + m;
    const int col     = colTile * 16 + m;

    // norm='both' source scaling, applied on the fly (row is same for both
    // lane halves, so this is one value per logical row)
    const float rs = __frsqrt_rn(fmaxf(deg_out[row], 1.0f));

    const float* xrow  = X + (long)row * F_INK + half * 2;        // A operand base
    const float* wbase = W + (long)(half * 2) * F_OUTK + col;     // B operand base

    v8f c = {};
    for (int k0 = 0; k0 < F_INK; k0 += 4) {
        // A: lane<16 holds K=k0,k0+1 ; lane>=16 holds K=k0+2,k0+3 (row-major,
        // consecutive -> single 8B load, 8B-aligned since k0%4==0, stride even)
        v2f a = *(const v2f*)(xrow + k0);
        a *= rs;
        // B: W is [300,64] row-major -> stride-64 between the two K rows
        v2f b;
        b.x = wbase[(long)k0 * F_OUTK];
        b.y = wbase[(long)(k0 + 1) * F_OUTK];
        // (neg_a, A, neg_b, B, c_mod, C, reuse_a, reuse_b)
        c = __builtin_amdgcn_wmma_f32_16x16x4_f32(false, a, false, b,
                                                  (short)0, c, false, false);
    }

    // D layout: VGPR i, lanes0-15 -> (M=i, N=lane); lanes16-31 -> (M=8+i, N=lane-16)
    float* hout = H + (long)(rowBase + half * 8) * F_OUTK + colTile * 16 + m;
#pragma unroll
    for (int i = 0; i < 8; ++i)
        hout[(long)i * F_OUTK] = c[i];
}

// ---------------------------------------------------------------------------
// Kernel 3: agg[dst] += H[src]  (16 threads per edge, float4 gather + 4
// f32 atomics each; agg is 33.5MB -> L2-resident atomics)
// ---------------------------------------------------------------------------
__global__ void gcn_scatter_kernel(const int* __restrict__ src,
                                   const int* __restrict__ dst,
                                   const float* __restrict__ H,
                                   float* __restrict__ agg) {
    long idx = (long)blockIdx.x * blockDim.x + threadIdx.x;   // E*16 threads
    int  e   = (int)(idx >> 4);
    int  q   = (int)(idx & 15);
    int  s   = src[e];
    int  d   = dst[e];
    float4 v = ((const float4*)(H + (long)s * F_OUTK))[q];
    float* out = agg + (long)d * F_OUTK + q * 4;
    atomicAdd(out + 0, v.x);
    atomicAdd(out + 1, v.y);
    atomicAdd(out + 2, v.z);
    atomicAdd(out + 3, v.w);
}

// ---------------------------------------------------------------------------
// Kernel 4: dest-normalize + bias + PReLU, fused with ragged mean-pool
// (exclude anchor) and anchor extraction. One block per subgraph.
// Thread layout: f = tid&63 (feature), g = tid>>6 (node group of 4).
// ---------------------------------------------------------------------------
__global__ void gcn_finalize_kernel(const float* __restrict__ agg,
                                    const float* __restrict__ deg_in,
                                    const float* __restrict__ bias,
                                    const float* __restrict__ prelu_a,
                                    float* __restrict__ pool,
                                    float* __restrict__ anchor) {
    __shared__ float red[256];
    const int  sg    = blockIdx.x;
    const int  f     = threadIdx.x & 63;
    const int  g     = threadIdx.x >> 6;
    const float alpha = prelu_a[0];
    const float bf    = bias[f];
    const long base   = (long)sg * S_SUB;

    float partial = 0.0f;
    for (int n = g; n < S_SUB - 1; n += 4) {
        long node = base + n;
        float ri = __frsqrt_rn(fmaxf(deg_in[node], 1.0f));
        float v  = agg[node * F_OUTK + f] * ri + bf;
        v = v > 0.0f ? v : alpha * v;
        partial += v;
    }

    if (g == 3) {  // anchor = last node of subgraph (excluded from pool)
        long node = base + S_SUB - 1;
        float ri = __frsqrt_rn(fmaxf(deg_in[node], 1.0f));
        float v  = agg[node * F_OUTK + f] * ri + bf;
        v = v > 0.0f ? v : alpha * v;
        anchor[(long)sg * F_OUTK + f] = v;
    }

    red[threadIdx.x] = partial;
    __syncthreads();
    if (g == 0) {
        float sum = red[f] + red[f + 64] + red[f + 128] + red[f + 192];
        pool[(long)sg * F_OUTK + f] = sum * (1.0f / (S_SUB - 1));
    }
}

// ---------------------------------------------------------------------------
extern "C" void kernel_launch(void* const* d_in, const int* in_sizes, int n_in,
                              void* d_out, int out_size, void* d_ws, size_t ws_size,
                              hipStream_t stream) {
    const float* in_feat  = (const float*)d_in[0];
    const float* W        = (const float*)d_in[1];
    const float* bias     = (const float*)d_in[2];
    const float* prelu_a  = (const float*)d_in[3];
    const int*   src      = (const int*)d_in[4];
    const int*   dst      = (const int*)d_in[5];

    float* ws      = (float*)d_ws;
    float* deg_out = ws;                               // N
    float* deg_in  = ws + N_NODES;                     // N
    float* H       = ws + 2L * N_NODES;                // N*64
    float* agg     = H + (long)N_NODES * F_OUTK;       // N*64

    float* pool   = (float*)d_out;                     // [1024,64]
    float* anchor = pool + (long)B_SUB * F_OUTK;       // [1024,64]

    // zero accumulators (capture-safe async memsets)
    hipMemsetAsync(deg_out, 0, 2L * N_NODES * sizeof(float), stream);
    hipMemsetAsync(agg, 0, (long)N_NODES * F_OUTK * sizeof(float), stream);

    gcn_degree_kernel<<<N_EDGE / 256, 256, 0, stream>>>(src, dst, deg_out, deg_in);
    gcn_gemm_wmma_kernel<<<N_NODES / 32, 256, 0, stream>>>(in_feat, W, deg_out, H);
    gcn_scatter_kernel<<<(int)((long)N_EDGE * 16 / 256), 256, 0, stream>>>(src, dst, H, agg);
    gcn_finalize_kernel<<<B_SUB, 256, 0, stream>>>(agg, deg_in, bias, prelu_a, pool, anchor);
}